// KabschRMSD_20633022890656
// MI455X (gfx1250) — compile-verified
//
#include <hip/hip_runtime.h>
#include <stdint.h>

typedef float v2f __attribute__((ext_vector_type(2)));
typedef float v8f __attribute__((ext_vector_type(8)));

#define BATCH   1024
#define NPTS    4096
#define CHUNK   1024
#define THREADS 256
#define NCHUNK  (NPTS / CHUNK)           // 4
#define F4_PER_CHUNK (CHUNK * 3 / 4)     // 768 float4 per tensor per chunk

// ---- CDNA5 async global->LDS copy (ASYNCcnt path, ISA ch.8) ----------------
__device__ __forceinline__ void async_copy_b128(uint32_t lds_byte_off,
                                                const void* gptr) {
    // GV mode: 64-bit global address in VGPR pair, LDS byte address in VGPR.
    asm volatile("global_load_async_to_lds_b128 %0, %1, off"
                 :: "v"(lds_byte_off), "v"((uint64_t)(uintptr_t)gptr)
                 : "memory");
}
__device__ __forceinline__ void wait_async0() {
    asm volatile("s_wait_asynccnt 0x0" ::: "memory");
}
__device__ __forceinline__ uint32_t lds_off(const void* shared_ptr) {
    // Low 32 bits of a generic pointer into LDS == LDS byte offset.
    return (uint32_t)(uintptr_t)shared_ptr;
}

// One workgroup per batch. Single pass over coords: all Kabsch moments
// accumulate through V_WMMA_F32_16X16X4_F32 (one WMMA per 4 points per wave),
// with double-buffered async global->LDS staging overlapping HBM with compute.
__global__ __launch_bounds__(THREADS) void kabsch_moments_kernel(
    const float* __restrict__ pred, const float* __restrict__ truec,
    float* __restrict__ rmsd_out)
{
    __shared__ float4 sbuf[2][2][F4_PER_CHUNK];  // [buffer][pred/true] : 48 KB
    __shared__ float  acc[64];                   // 7x7 used, stride 8

    const int b    = blockIdx.x;
    const int t    = threadIdx.x;
    const int wave = t >> 5;
    const int lane = t & 31;
    const int m    = lane & 15;   // row of A / col of B this lane feeds
    const int hi   = lane >> 4;   // selects K=0,1 vs K=2,3

    if (t < 64) acc[t] = 0.0f;

    // Branchless operand construction:
    //   row m<3  -> pred coord m (A) / true coord m (B)
    //   row m==3 -> 1.0
    //   row 4..6 -> true coord m-4 (A) / pred coord m-4 (B)
    //   row >=7  -> 0.0
    const float scale = (m == 3 || m > 6) ? 0.0f : 1.0f;
    const float plus  = (m == 3) ? 1.0f : 0.0f;
    int off = (m < 3) ? m : (m - 4);
    off = (off < 0) ? 0 : ((off > 2) ? 2 : off);
    const bool mIsPred = (m < 3);

    v8f c = {};

    const float4* gp = (const float4*)(pred  + (size_t)b * NPTS * 3);
    const float4* gt = (const float4*)(truec + (size_t)b * NPTS * 3);

    // Issue async copies of one chunk into buffer `buf`.
    auto issue_chunk = [&](int ch, int buf) {
        const int f4base = ch * F4_PER_CHUNK;
#pragma unroll
        for (int i = 0; i < F4_PER_CHUNK / THREADS; ++i) {   // 3 per tensor
            const int idx = t + i * THREADS;
            async_copy_b128(lds_off(&sbuf[buf][0][idx]), gp + f4base + idx);
            async_copy_b128(lds_off(&sbuf[buf][1][idx]), gt + f4base + idx);
        }
    };

    // Prologue: chunk 0 -> buffer 0.
    issue_chunk(0, 0);
    wait_async0();
    __syncthreads();

    for (int ch = 0; ch < NCHUNK; ++ch) {
        const int cur = ch & 1;
        if (ch + 1 < NCHUNK) issue_chunk(ch + 1, (ch + 1) & 1);  // overlap

        const float* sP = (const float*)sbuf[cur][0];
        const float* sT = (const float*)sbuf[cur][1];
        const float* aB = mIsPred ? sP : sT;
        const float* bB = mIsPred ? sT : sP;

        // This wave covers points [wave*128, wave*128+128) of the chunk.
        const int pbase = wave * (CHUNK / 8);
#pragma unroll 4
        for (int g = 0; g < (CHUNK / 8) / 4; ++g) {
            const int p0 = pbase + g * 4 + 2 * hi;  // lane's K pair: p0, p0+1
            const int i0 = p0 * 3 + off;
            float a0 = aB[i0]     * scale + plus;
            float a1 = aB[i0 + 3] * scale + plus;
            float b0 = bB[i0]     * scale + plus;
            float b1 = bB[i0 + 3] * scale + plus;
            v2f av = {a0, a1};
            v2f bv = {b0, b1};
            c = __builtin_amdgcn_wmma_f32_16x16x4_f32(
                    false, av, false, bv, (short)0, c, false, false);
        }

        wait_async0();       // this wave's copies into the next buffer done
        __syncthreads();     // everyone done reading cur + next buffer ready
    }

    // Merge 8 wave accumulators. Lanes 0-15 hold rows M=0..7 in c[0..7];
    // only M<7, N<7 carry payload.
    if (lane < 7) {
#pragma unroll
        for (int v = 0; v < 7; ++v)
            atomicAdd(&acc[v * 8 + lane], c[v]);
    }
    __syncthreads();

    if (t == 0) {
        float M[3][3], sp[3], st[3];
        for (int i = 0; i < 3; ++i)
            for (int j = 0; j < 3; ++j) M[i][j] = acc[i * 8 + j];
        for (int i = 0; i < 3; ++i) sp[i] = acc[i * 8 + 3];
        for (int j = 0; j < 3; ++j) st[j] = acc[24 + j];
        const float spp = acc[4] + acc[8 + 5] + acc[16 + 6];
        const float stt = acc[32] + acc[40 + 1] + acc[48 + 2];
        const float invN = 1.0f / (float)NPTS;

        float A[3][3];
        for (int i = 0; i < 3; ++i)
            for (int j = 0; j < 3; ++j)
                A[i][j] = M[i][j] - sp[i] * st[j] * invN;
        const float Epp = spp - (sp[0]*sp[0] + sp[1]*sp[1] + sp[2]*sp[2]) * invN;
        const float Ett = stt - (st[0]*st[0] + st[1]*st[1] + st[2]*st[2]) * invN;

        const float detA =
            A[0][0]*(A[1][1]*A[2][2] - A[1][2]*A[2][1]) -
            A[0][1]*(A[1][0]*A[2][2] - A[1][2]*A[2][0]) +
            A[0][2]*(A[1][0]*A[2][1] - A[1][1]*A[2][0]);

        // G = A^T A, symmetric PSD; closed-form eigenvalues.
        float G[3][3];
        for (int i = 0; i < 3; ++i)
            for (int j = 0; j < 3; ++j) {
                float s = 0.0f;
                for (int k = 0; k < 3; ++k) s += A[k][i] * A[k][j];
                G[i][j] = s;
            }
        const float q = (G[0][0] + G[1][1] + G[2][2]) * (1.0f / 3.0f);
        const float p2 = (G[0][0]-q)*(G[0][0]-q) + (G[1][1]-q)*(G[1][1]-q) +
                         (G[2][2]-q)*(G[2][2]-q) +
                         2.0f*(G[0][1]*G[0][1] + G[0][2]*G[0][2] + G[1][2]*G[1][2]);
        const float p = sqrtf(fmaxf(p2 * (1.0f / 6.0f), 0.0f));
        float l1, l2, l3;
        if (p < 1e-20f) {
            l1 = l2 = l3 = q;
        } else {
            const float ip = 1.0f / p;
            float Bm[3][3];
            for (int i = 0; i < 3; ++i)
                for (int j = 0; j < 3; ++j)
                    Bm[i][j] = (G[i][j] - (i == j ? q : 0.0f)) * ip;
            const float detB =
                Bm[0][0]*(Bm[1][1]*Bm[2][2] - Bm[1][2]*Bm[2][1]) -
                Bm[0][1]*(Bm[1][0]*Bm[2][2] - Bm[1][2]*Bm[2][0]) +
                Bm[0][2]*(Bm[1][0]*Bm[2][1] - Bm[1][1]*Bm[2][0]);
            float r = fminf(1.0f, fmaxf(-1.0f, detB * 0.5f));
            const float phi = acosf(r) * (1.0f / 3.0f);
            l1 = q + 2.0f * p * cosf(phi);
            l3 = q + 2.0f * p * cosf(phi + 2.0943951023931953f);
            l2 = 3.0f * q - l1 - l3;
        }
        const float s1 = sqrtf(fmaxf(l1, 0.0f));
        const float s2 = sqrtf(fmaxf(l2, 0.0f));
        const float s3 = sqrtf(fmaxf(l3, 0.0f));
        const float d  = (detA < 0.0f) ? -1.0f : 1.0f;
        const float trS = s1 + s2 + d * s3;
        const float msd = fmaxf(Epp + Ett - 2.0f * trS, 0.0f) * invN;
        rmsd_out[b] = sqrtf(msd);
    }
}

// Deterministic tree reduction: mean over 1024 per-batch RMSDs.
__global__ __launch_bounds__(256) void mean_reduce_kernel(
    const float* __restrict__ rmsd, float* __restrict__ out)
{
    __shared__ float red[256];
    const int t = threadIdx.x;
    float s = 0.0f;
    for (int i = t; i < BATCH; i += 256) s += rmsd[i];
    red[t] = s;
    __syncthreads();
    for (int w = 128; w > 0; w >>= 1) {
        if (t < w) red[t] += red[t + w];
        __syncthreads();
    }
    if (t == 0) out[0] = red[0] * (1.0f / (float)BATCH);
}

extern "C" void kernel_launch(void* const* d_in, const int* in_sizes, int n_in,
                              void* d_out, int out_size, void* d_ws, size_t ws_size,
                              hipStream_t stream) {
    const float* pred  = (const float*)d_in[0];
    const float* truec = (const float*)d_in[1];
    float* rmsd_ws = (float*)d_ws;   // 1024 floats of scratch

    kabsch_moments_kernel<<<BATCH, THREADS, 0, stream>>>(pred, truec, rmsd_ws);
    mean_reduce_kernel<<<1, 256, 0, stream>>>(rmsd_ws, (float*)d_out);
}